// CopyMechModule_15814069584249
// MI455X (gfx1250) — compile-verified
//
#include <hip/hip_runtime.h>

typedef float v2f  __attribute__((ext_vector_type(2)));
typedef float v8f  __attribute__((ext_vector_type(8)));
typedef unsigned int u32x4 __attribute__((ext_vector_type(4)));
typedef int   i32x4 __attribute__((ext_vector_type(4)));
typedef int   i32x8 __attribute__((ext_vector_type(8)));

#define B_SZ   8
#define T_SZ   512
#define S_SZ   512
#define H_SZ   768
#define VOCAB  32110
#define VOCAB_PAD 32112          // pad to float4 multiple for LDS zeroing
#define ROWS   (B_SZ * T_SZ)     // 4096 (b,t) rows

// ---------------------------------------------------------------------------
// Kernel 1: p_gen = sigmoid(concat(dec,seq) @ w + b) via V_WMMA_F32_16X16X4_F32
// One wave produces 16 rows. B-matrix columns all hold w[k], so every column
// of D is the desired dot product; lanes 0 and 16 own rows 0-7 / 8-15 in their
// 8 accumulator VGPRs (documented 16x16 f32 C/D layout). The concat is handled
// as two contiguous K-halves (no per-iteration pointer selects).
// ---------------------------------------------------------------------------
__global__ __launch_bounds__(256) void pgen_wmma_kernel(
    const float* __restrict__ dec, const float* __restrict__ seq,
    const float* __restrict__ w,   const float* __restrict__ bptr,
    float* __restrict__ out)
{
  const int lane = threadIdx.x & 31;
  const int wave = threadIdx.x >> 5;
  const int m    = lane & 15;     // A-matrix row (M)
  const int sel  = lane >> 4;     // 0 -> K pair (0,1), 1 -> K pair (2,3)
  const int rowBase = (blockIdx.x * 8 + wave) * 16;
  const int row     = rowBase + m;

  const float* decRow = dec + (size_t)row * H_SZ + 2 * sel;
  const float* seqRow = seq + (size_t)row * H_SZ + 2 * sel;
  const float* w0     = w + 2 * sel;          // first K-half weights
  const float* w1     = w + H_SZ + 2 * sel;   // second K-half weights

  v8f acc0 = {0.f, 0.f, 0.f, 0.f, 0.f, 0.f, 0.f, 0.f};
  v8f acc1 = acc0;

  // K-half 1: decoder embeds (k = 0..767), two WMMAs per iteration
  for (int kb = 0; kb < H_SZ; kb += 8) {
    v2f a0 = *(const v2f*)(decRow + kb);
    v2f a1 = *(const v2f*)(decRow + kb + 4);
    v2f b0 = *(const v2f*)(w0 + kb);
    v2f b1 = *(const v2f*)(w0 + kb + 4);
    acc0 = __builtin_amdgcn_wmma_f32_16x16x4_f32(false, a0, false, b0,
                                                 (short)0, acc0, false, false);
    acc1 = __builtin_amdgcn_wmma_f32_16x16x4_f32(false, a1, false, b1,
                                                 (short)0, acc1, false, false);
  }
  // K-half 2: sequence output (k = 768..1535)
  for (int kb = 0; kb < H_SZ; kb += 8) {
    v2f a0 = *(const v2f*)(seqRow + kb);
    v2f a1 = *(const v2f*)(seqRow + kb + 4);
    v2f b0 = *(const v2f*)(w1 + kb);
    v2f b1 = *(const v2f*)(w1 + kb + 4);
    acc0 = __builtin_amdgcn_wmma_f32_16x16x4_f32(false, a0, false, b0,
                                                 (short)0, acc0, false, false);
    acc1 = __builtin_amdgcn_wmma_f32_16x16x4_f32(false, a1, false, b1,
                                                 (short)0, acc1, false, false);
  }
  v8f acc = acc0 + acc1;

  if (m == 0) {                                // lanes 0 and 16 hold the results
    const float bias = bptr[0];
    const int base = rowBase + sel * 8;
    #pragma unroll
    for (int i = 0; i < 8; ++i) {
      float v = acc[i] + bias;
      out[base + i] = 1.0f / (1.0f + __expf(-v));
    }
  }
}

// ---------------------------------------------------------------------------
// Kernel 2: logits[b,t,:] = segment-sum of attn[b,t,:] by ids[b,:].
// One workgroup per (b,t) row. Row built in LDS (128.4 KB of the 320 KB WGP
// LDS; 2 workgroups co-resident), scattered with ds_add_f32, then DMA'd to
// HBM with a single TENSOR_STORE_FROM_LDS (TDM) — one write per output byte.
// ---------------------------------------------------------------------------
__global__ __launch_bounds__(512) void copy_logits_kernel(
    const float* __restrict__ attn, const int* __restrict__ ids,
    float* __restrict__ logits)
{
  extern __shared__ float lds[];
  const int tid = threadIdx.x;
  const int bt  = blockIdx.x;          // b*T + t
  const int b   = bt >> 9;             // / T_SZ

  // Zero the vocab row in LDS (float4 -> ds_store_b128)
  float4* lds4 = (float4*)lds;
  const float4 z = make_float4(0.f, 0.f, 0.f, 0.f);
  for (int i = tid; i < VOCAB_PAD / 4; i += 512) lds4[i] = z;
  __syncthreads();

  // Scatter-add: 512 threads, one source element each (ds_add_f32, no return)
  {
    const int   s  = tid;
    const int   id = ids[b * S_SZ + s];
    const float a  = attn[(size_t)bt * S_SZ + s];
    atomicAdd(&lds[id], a);
  }
  __syncthreads();

  // One TDM descriptor: 2D tile (32110 x 1) of 4-byte elements, LDS -> global.
  if (__builtin_amdgcn_readfirstlane(threadIdx.x) == 0) {   // wave 0 only
    unsigned long long ga =
        (unsigned long long)(logits + (size_t)bt * VOCAB);

    u32x4 g0;
    g0[0] = 1u;                                   // count=1, user mode, no gather
    g0[1] = 0u;                                   // lds_addr = 0 (dynamic LDS base)
    g0[2] = (unsigned)(ga & 0xFFFFFFFFu);         // global_addr[31:0]
    g0[3] = (unsigned)((ga >> 32) & 0x1FFFFFFu)   // global_addr[56:32]
          | (2u << 30);                           // type = 2 ("image")

    i32x8 g1;
    g1[0] = (int)(2u << 16);                      // data_size=4B; mask=0; no pad
    g1[1] = (int)(((unsigned)VOCAB & 0xFFFFu) << 16); // tensor_dim0[15:0]
    g1[2] = (int)(1u << 16);                      // tensor_dim0[31:16]=0 | tensor_dim1=1
    g1[3] = (int)(((unsigned)VOCAB & 0xFFFFu) << 16); // tile_dim0 = 32110
    g1[4] = 1;                                    // tile_dim1=1, tile_dim2=0
    g1[5] = VOCAB;                                // tensor_dim0_stride[31:0]
    g1[6] = (int)(((unsigned)VOCAB & 0xFFFFu) << 16); // stride0 hi=0 | stride1 lo
    g1[7] = 0;                                    // stride1 hi

    i32x4 g2;
    g2[0] = 1;                                    // tensor_dim2 = 1
    g2[1] = 1;                                    // tensor_dim3 = 1 (iterate off)
    g2[2] = 0;                                    // tensor_dim2_stride lo
    g2[3] = 0;                                    // stride hi | tile_dim3 = 0

    i32x4 g3;
    g3[0] = 0;                                    // tensor_dim3_stride lo
    g3[1] = (int)(1u << 16);                      // tensor_dim4 = 1
    g3[2] = 0;                                    // tile_dim4 = 0
    g3[3] = 0;

    i32x8 gx = {0, 0, 0, 0, 0, 0, 0, 0};          // 6-arg toolchain: extra group

    __builtin_amdgcn_tensor_store_from_lds(g0, g1, g2, g3, gx, 0);
    __builtin_amdgcn_s_wait_tensorcnt(0);         // hold LDS until DMA drains
  }
}

// ---------------------------------------------------------------------------
extern "C" void kernel_launch(void* const* d_in, const int* in_sizes, int n_in,
                              void* d_out, int out_size, void* d_ws, size_t ws_size,
                              hipStream_t stream) {
  const float* dec  = (const float*)d_in[0];   // [B,T,H]
  const float* seq  = (const float*)d_in[1];   // [B,T,H]
  const float* attn = (const float*)d_in[2];   // [B,T,S]
  const int*   ids  = (const int*)  d_in[3];   // [B,S]
  const float* w    = (const float*)d_in[4];   // [2H,1]
  const float* bp   = (const float*)d_in[5];   // [1]

  float* pgen   = (float*)d_out;               // [B,T,1] = 4096 floats
  float* logits = (float*)d_out + ROWS;        // [B,T,VOCAB]

  pgen_wmma_kernel<<<ROWS / 128, 256, 0, stream>>>(dec, seq, w, bp, pgen);
  copy_logits_kernel<<<ROWS, 512, VOCAB_PAD * sizeof(float), stream>>>(
      attn, ids, logits);
}